// Quartic_63247688401600
// MI455X (gfx1250) — compile-verified
//
#include <hip/hip_runtime.h>
#include <hip/hip_bf16.h>

#define TT    32768   // 32*32*32 type-triple table size
#define NTHR  256     // threads per block (pairs per tile)
#define TELEM 512     // elements per tile (2 per thread)

typedef __attribute__((address_space(1))) int g1_int;
typedef __attribute__((address_space(3))) int l3_int;

// Confirmed-working async global->LDS B32 copy (round 2/3 probes).
__device__ __forceinline__ void async_copy_b32(const int* __restrict__ gsrc, int* lds_dst) {
#if __has_builtin(__builtin_amdgcn_global_load_async_to_lds_b32)
    __builtin_amdgcn_global_load_async_to_lds_b32(
        (g1_int*)(size_t)(const void*)gsrc, (l3_int*)lds_dst, /*offset=*/0, /*cpol=*/0);
#else
    *lds_dst = *gsrc;   // synchronous fallback
#endif
}

// 8 async ops per wave per tile (4 arrays x 2 elements).
__device__ __forceinline__ void wait_asynccnt_le8() {
#if __has_builtin(__builtin_amdgcn_s_wait_asynccnt)
    __builtin_amdgcn_s_wait_asynccnt(8);
#else
    asm volatile("s_wait_asynccnt 0x8" ::: "memory");
#endif
}

__global__ void zero_out_kernel(float* __restrict__ out, int n) {
    int i = blockIdx.x * blockDim.x + threadIdx.x;
    if (i < n) out[i] = 0.0f;
}

__global__ __launch_bounds__(NTHR) void quartic_angle_kernel(
    const float* __restrict__ pos,    // (N_ATOMS, 3)
    const int*   __restrict__ map0,   // (N_INT)
    const int*   __restrict__ map1,   // (N_INT)
    const int*   __restrict__ map2,   // (N_INT)
    const int*   __restrict__ atype,  // (N_ATOMS)
    const int*   __restrict__ batch,  // (N_INT), sorted
    const float* __restrict__ ks,     // (3, 32,32,32)
    const float* __restrict__ x0s,    // (3, 32,32,32)
    const float* __restrict__ v0,     // (32,32,32)
    float*       __restrict__ out,    // (N_MOL)
    int n, int ntiles)
{
    // Double-buffered staging of the streaming arrays (16 KB LDS).
    __shared__ int s0[2][TELEM], s1[2][TELEM], s2[2][TELEM], sb[2][TELEM];

    const int tid  = threadIdx.x;
    const int lane = tid & 31;

    // Stage one 512-element tile into buffer `bufi`. Per-element clamping at
    // B32 granularity: tail slots get in-bounds garbage that `valid` ignores.
    // Every wave issues exactly 8 async ops per call.
    auto stage = [&](int t, int bufi) {
        const int e0 = t * TELEM + 2 * tid;
        const int g0 = (e0     < n) ? e0     : (n - 1);
        const int g1 = (e0 + 1 < n) ? e0 + 1 : (n - 1);
        async_copy_b32(map0  + g0, &s0[bufi][2 * tid]);
        async_copy_b32(map0  + g1, &s0[bufi][2 * tid + 1]);
        async_copy_b32(map1  + g0, &s1[bufi][2 * tid]);
        async_copy_b32(map1  + g1, &s1[bufi][2 * tid + 1]);
        async_copy_b32(map2  + g0, &s2[bufi][2 * tid]);
        async_copy_b32(map2  + g1, &s2[bufi][2 * tid + 1]);
        async_copy_b32(batch + g0, &sb[bufi][2 * tid]);
        async_copy_b32(batch + g1, &sb[bufi][2 * tid + 1]);
    };

    // One interaction's energy: angle + per-type-triple quartic polynomial.
    auto energy = [&](int a0, int a1, int a2) -> float {
        const float* p0 = pos + 3 * a0;
        const float* p1 = pos + 3 * a1;
        const float* p2 = pos + 3 * a2;
        const float p1x = p1[0], p1y = p1[1], p1z = p1[2];
        const float r1x = p0[0] - p1x, r1y = p0[1] - p1y, r1z = p0[2] - p1z;
        const float r2x = p2[0] - p1x, r2y = p2[1] - p1y, r2z = p2[2] - p1z;

        const float cx = r1y * r2z - r1z * r2y;
        const float cy = r1z * r2x - r1x * r2z;
        const float cz = r1x * r2y - r1y * r2x;
        const float sin_t = sqrtf(cx * cx + cy * cy + cz * cz);
        const float cos_t = r1x * r2x + r1y * r2y + r1z * r2z;
        const float x = atan2f(sin_t, cos_t);

        const int t0 = atype[a0];
        const int t1 = atype[a1];
        const int t2 = atype[a2];
        const int idx = (t0 * 32 + t1) * 32 + t2;

        const float d0 = x - x0s[idx];
        const float d1 = x - x0s[TT + idx];
        const float d2 = x - x0s[2 * TT + idx];
        const float d2sq = d2 * d2;
        return ks[idx] * d0 * d0
             + ks[TT + idx] * d1 * d1 * d1
             + ks[2 * TT + idx] * d2sq * d2sq
             + v0[idx];
    };

    int t = blockIdx.x;
    if (t >= ntiles) return;

    stage(t, 0);          // prologue
    int buf = 0;

    for (; t < ntiles; t += gridDim.x) {
        // Stage next tile (clamped on last iteration: uniform 8 asyncs/wave).
        const int tnext  = t + gridDim.x;
        const int tstage = (tnext < ntiles) ? tnext : t;
        stage(tstage, buf ^ 1);

        // Oldest 8 async ops (current tile) complete when cnt <= 8.
        wait_asynccnt_le8();

        const int  i0     = t * TELEM + 2 * tid;
        const int  i1     = i0 + 1;
        const bool valid0 = (i0 < n);
        const bool valid1 = (i1 < n);

        // Carried (molecule id, partial sum) for the cross-lane scan.
        int   b = -1;
        float v = 0.0f;

        if (valid0) {
            const int   b0 = sb[buf][2 * tid];
            const float v0e = energy(s0[buf][2 * tid],
                                     s1[buf][2 * tid],
                                     s2[buf][2 * tid]);
            if (valid1) {
                const int   b1 = sb[buf][2 * tid + 1];
                const float v1e = energy(s0[buf][2 * tid + 1],
                                         s1[buf][2 * tid + 1],
                                         s2[buf][2 * tid + 1]);
                if (b0 == b1) {               // common: pair in same molecule
                    b = b0; v = v0e + v1e;
                } else {                      // rare: pair straddles boundary
                    atomicAdd(out + b0, v0e); // flush first element directly
                    b = b1; v = v1e;
                }
            } else {
                b = b0; v = v0e;
            }
        }

        // Wave32 segmented suffix-sum over carried runs (batch is sorted).
        #pragma unroll
        for (int off = 1; off < 32; off <<= 1) {
            const float vo = __shfl_down(v, off, 32);
            const int   bo = __shfl_down(b, off, 32);
            if ((lane + off < 32) && (bo == b)) v += vo;
        }

        const int  bprev = __shfl_up(b, 1, 32);
        const bool head  = (lane == 0) || (bprev != b);

        if ((b >= 0) && head) {
            atomicAdd(out + b, v);  // ~1 atomic per wave
        }

        buf ^= 1;
    }
}

extern "C" void kernel_launch(void* const* d_in, const int* in_sizes, int n_in,
                              void* d_out, int out_size, void* d_ws, size_t ws_size,
                              hipStream_t stream) {
    const float* pos     = (const float*)d_in[0];
    const int*   mapping = (const int*)  d_in[1];  // (3, N_INT) row-major
    const int*   atype   = (const int*)  d_in[2];
    const int*   batch   = (const int*)  d_in[3];
    const float* ks      = (const float*)d_in[4];  // (3, 32,32,32)
    const float* x0s     = (const float*)d_in[5];  // (3, 32,32,32)
    const float* v0      = (const float*)d_in[6];  // (32,32,32)
    float*       out     = (float*)d_out;

    const int n = in_sizes[3];  // N_INT (length of mapping_batch)
    const int* map0 = mapping;
    const int* map1 = mapping + (size_t)n;
    const int* map2 = mapping + 2 * (size_t)n;

    // d_out is accumulated via atomics: zero it every launch (graph-safe).
    zero_out_kernel<<<(out_size + 255) / 256, 256, 0, stream>>>(out, out_size);

    const int ntiles = (n + TELEM - 1) / TELEM;
    int blocks = 2048;                 // persistent blocks; grid-stride tiles
    if (blocks > ntiles) blocks = ntiles;

    quartic_angle_kernel<<<blocks, NTHR, 0, stream>>>(
        pos, map0, map1, map2, atype, batch, ks, x0s, v0, out, n, ntiles);
}